// GNNEncoder_36756330119414
// MI455X (gfx1250) — compile-verified
//
#include <hip/hip_runtime.h>
#include <math.h>

typedef __attribute__((ext_vector_type(2))) float v2f;
typedef __attribute__((ext_vector_type(8))) float v8f;

#define NNODES 100000
#define NEDGES 1000000
#define DIM 64
#define NEG_SLOPE 0.2f
#define NSTRIPS (NNODES / 16)      // 6250 exact
#define NDSTRIPS (NSTRIPS / 2)     // 3125 double-strips (32 rows each)

static __device__ __forceinline__ v8f wmma4(v2f a, v2f b, v8f c) {
    return __builtin_amdgcn_wmma_f32_16x16x4_f32(false, a, false, b, (short)0, c, false, false);
}

// ---------- float atomic max via signed/unsigned integer ordering trick ----------
static __device__ __forceinline__ void atomic_max_f32(float* addr, float v) {
    if (v >= 0.0f) atomicMax((int*)addr, __float_as_int(v));
    else           atomicMin((unsigned int*)addr, __float_as_uint(v));
}

// ---------- fold attention vectors: v = W @ att  (64 threads) ----------
__global__ void gat_prep(const float* __restrict__ Wsrc, const float* __restrict__ Wdst,
                         const float* __restrict__ att_s, const float* __restrict__ att_d,
                         float* __restrict__ vsrc, float* __restrict__ vdst) {
    int d = threadIdx.x;
    float s = 0.f, t = 0.f;
    #pragma unroll
    for (int h = 0; h < DIM; ++h) {
        s += Wsrc[d * DIM + h] * att_s[h];
        t += Wdst[d * DIM + h] * att_d[h];
    }
    vsrc[d] = s;
    vdst[d] = t;
}

// ---------- per-call workspace init (harness does not re-zero) ----------
__global__ void gat_init(float* __restrict__ segmax, float* __restrict__ denom,
                         float* __restrict__ agg) {
    size_t i = (size_t)blockIdx.x * blockDim.x + threadIdx.x;
    if (i < (size_t)NNODES * DIM) agg[i] = 0.f;
    if (i < NNODES) { segmax[i] = -INFINITY; denom[i] = 0.f; }
}

// ---------- per-node attention scores: a = x . v ----------
__global__ void gat_scores(const float* __restrict__ x, const float* __restrict__ vsrc,
                           const float* __restrict__ vdst,
                           float* __restrict__ a_src, float* __restrict__ a_dst) {
    int n = blockIdx.x * blockDim.x + threadIdx.x;
    if (n >= NNODES) return;
    const float4* xr = (const float4*)(x + (size_t)n * DIM);
    const float4* vs = (const float4*)vsrc;
    const float4* vd = (const float4*)vdst;
    float s = 0.f, t = 0.f;
    #pragma unroll
    for (int q = 0; q < DIM / 4; ++q) {
        float4 v = xr[q], a = vs[q], b = vd[q];
        s += v.x * a.x + v.y * a.y + v.z * a.z + v.w * a.w;
        t += v.x * b.x + v.y * b.y + v.z * b.z + v.w * b.w;
    }
    a_src[n] = s;
    a_dst[n] = t;
}

// ---------- edge pass 1: segment max of leaky-relu logits ----------
__global__ void gat_edge_max(const int* __restrict__ ei, const float* __restrict__ a_src,
                             const float* __restrict__ a_dst, float* __restrict__ segmax) {
    int e = blockIdx.x * blockDim.x + threadIdx.x;
    if (e >= NEDGES) return;
    int s = ei[e], t = ei[NEDGES + e];
    float v = a_src[s] + a_dst[t];
    v = (v > 0.f) ? v : NEG_SLOPE * v;
    atomic_max_f32(&segmax[t], v);
}

// ---------- edge pass 2: exp(e - max) and segment sum ----------
__global__ void gat_edge_exp(const int* __restrict__ ei, const float* __restrict__ a_src,
                             const float* __restrict__ a_dst, const float* __restrict__ segmax,
                             float* __restrict__ expw, float* __restrict__ denom) {
    int e = blockIdx.x * blockDim.x + threadIdx.x;
    if (e >= NEDGES) return;
    int s = ei[e], t = ei[NEDGES + e];
    float v = a_src[s] + a_dst[t];
    v = (v > 0.f) ? v : NEG_SLOPE * v;
    float w = expf(v - segmax[t]);
    expw[e] = w;
    atomicAdd(&denom[t], w);
}

// ---------- edge pass 3: wave-per-edge weighted scatter of h_src rows ----------
// Next-edge gather prefetched one iteration ahead (global_prefetch_b8) to hide
// the dependent index->row latency behind this edge's atomics.
__global__ void gat_edge_aggr(const int* __restrict__ ei, const float* __restrict__ expw,
                              const float* __restrict__ denom, const float* __restrict__ hsrc,
                              float* __restrict__ agg) {
    int lane  = threadIdx.x & 31;
    int wave  = (blockIdx.x * blockDim.x + threadIdx.x) >> 5;
    int nwave = (gridDim.x * blockDim.x) >> 5;
    for (int e = wave; e < NEDGES; e += nwave) {
        int en = e + nwave;
        if (en < NEDGES) {
            int sn = ei[en];
            __builtin_prefetch(hsrc + (size_t)sn * DIM + lane * 2, 0, 3);
        }
        int s = ei[e], t = ei[NEDGES + e];
        float alpha = expw[e] / (denom[t] + 1e-16f);
        const float2 hv = *(const float2*)(hsrc + (size_t)s * DIM + lane * 2);
        atomicAdd(&agg[(size_t)t * DIM + lane * 2 + 0], alpha * hv.x);
        atomicAdd(&agg[(size_t)t * DIM + lane * 2 + 1], alpha * hv.y);
    }
}

// ---------- fp32 WMMA GEMM: C[M x 64] = act(A[M x 64] (+bias_in) @ B[64 x 64] (+bias_out))
// Wave computes a 32-row double-strip (2 x 4 tiles of 16x16); K swept in steps
// of 4 via V_WMMA_F32_16X16X4_F32. W pre-swizzled into LDS as contiguous
// (K,K+1) pairs per half-wave so each B operand is ONE aligned ds_load_b64
// (no cross-register packing movs), and every B pair feeds 2 wmma.
template <bool FINAL>
__launch_bounds__(256)
__global__ void gat_gemm(const float* __restrict__ A, const float* __restrict__ B,
                         const float* __restrict__ bias_in, const float* __restrict__ bias_out,
                         float* __restrict__ C) {
    // Blds[kk*128 + half*64 + c] = { B[4kk+2half][c], B[4kk+2half+1][c] }
    __shared__ v2f Blds[16 * 2 * DIM];   // 16 KB
    for (int p = threadIdx.x; p < 16 * 2 * DIM; p += blockDim.x) {
        int kk = p >> 7, half = (p >> 6) & 1, c = p & 63;
        int k = kk * 4 + half * 2;
        v2f pr;
        pr.x = B[k * DIM + c];
        pr.y = B[(k + 1) * DIM + c];
        Blds[p] = pr;
    }
    __syncthreads();

    const int lane = threadIdx.x & 31;
    const int col  = lane & 15;     // tile column / A-tile row
    const int hi   = lane >> 4;     // half-wave: A layout lanes 16-31 carry K+2,K+3
    const int koff = hi * 2;

    int dstrip = blockIdx.x * (blockDim.x >> 5) + (threadIdx.x >> 5);
    if (dstrip >= NDSTRIPS) return;
    const int m0 = dstrip * 32;
    const float* Arow0 = A + (size_t)(m0 + col) * DIM;
    const float* Arow1 = A + (size_t)(m0 + 16 + col) * DIM;

    v8f c00 = {}, c01 = {}, c02 = {}, c03 = {};
    v8f c10 = {}, c11 = {}, c12 = {}, c13 = {};
    #pragma unroll
    for (int kk = 0; kk < 16; ++kk) {
        const int k0 = kk * 4;
        v2f a0 = *(const v2f*)(Arow0 + k0 + koff);
        v2f a1 = *(const v2f*)(Arow1 + k0 + koff);
        if (FINAL) {  // reference adds bias to aggregate BEFORE the final linear
            v2f bi = *(const v2f*)(bias_in + k0 + koff);
            a0 += bi;
            a1 += bi;
        }
        const v2f* Bp = &Blds[kk * 128 + hi * 64 + col];
        v2f b0 = Bp[0], b1 = Bp[16], b2 = Bp[32], b3 = Bp[48];
        c00 = wmma4(a0, b0, c00);  c10 = wmma4(a1, b0, c10);
        c01 = wmma4(a0, b1, c01);  c11 = wmma4(a1, b1, c11);
        c02 = wmma4(a0, b2, c02);  c12 = wmma4(a1, b2, c12);
        c03 = wmma4(a0, b3, c03);  c13 = wmma4(a1, b3, c13);
    }

    // C/D layout: VGPR r holds row (strip_base + hi*8 + r), column n*16 + col
    v8f acc0[4] = {c00, c01, c02, c03};
    v8f acc1[4] = {c10, c11, c12, c13};
    #pragma unroll
    for (int n = 0; n < 4; ++n) {
        #pragma unroll
        for (int r = 0; r < 8; ++r) {
            float v0 = acc0[n][r];
            float v1 = acc1[n][r];
            if (FINAL) {
                float bo = bias_out[n * 16 + col];
                v0 = fmaxf(v0 + bo, 0.f);
                v1 = fmaxf(v1 + bo, 0.f);
            }
            C[(size_t)(m0 + hi * 8 + r) * DIM + n * 16 + col]      = v0;
            C[(size_t)(m0 + 16 + hi * 8 + r) * DIM + n * 16 + col] = v1;
        }
    }
}

extern "C" void kernel_launch(void* const* d_in, const int* in_sizes, int n_in,
                              void* d_out, int out_size, void* d_ws, size_t ws_size,
                              hipStream_t stream) {
    const float* x     = (const float*)d_in[0];
    const int*   ei    = (const int*)d_in[1];
    const float* W_src = (const float*)d_in[2];
    const float* W_dst = (const float*)d_in[3];
    const float* att_s = (const float*)d_in[4];
    const float* att_d = (const float*)d_in[5];
    const float* bias  = (const float*)d_in[6];
    const float* W_lin = (const float*)d_in[7];
    const float* b_lin = (const float*)d_in[8];
    float* out = (float*)d_out;

    // workspace carve-up (~56.8 MB of f32)
    float* p = (float*)d_ws;
    float* vsrc   = p; p += 64;
    float* vdst   = p; p += 64;
    float* a_src  = p; p += NNODES;
    float* a_dst  = p; p += NNODES;
    float* segmax = p; p += NNODES;
    float* denom  = p; p += NNODES;
    float* expw   = p; p += NEDGES;
    float* hsrc   = p; p += (size_t)NNODES * DIM;
    float* agg    = p; p += (size_t)NNODES * DIM;

    const int T = 256;
    const int gemm_blocks = (NDSTRIPS + (T / 32) - 1) / (T / 32);   // 8 double-strips/block

    gat_prep<<<1, 64, 0, stream>>>(W_src, W_dst, att_s, att_d, vsrc, vdst);
    gat_init<<<((size_t)NNODES * DIM + T - 1) / T, T, 0, stream>>>(segmax, denom, agg);
    gat_gemm<false><<<gemm_blocks, T, 0, stream>>>(x, W_src, nullptr, nullptr, hsrc);
    gat_scores<<<(NNODES + T - 1) / T, T, 0, stream>>>(x, vsrc, vdst, a_src, a_dst);
    gat_edge_max<<<(NEDGES + T - 1) / T, T, 0, stream>>>(ei, a_src, a_dst, segmax);
    gat_edge_exp<<<(NEDGES + T - 1) / T, T, 0, stream>>>(ei, a_src, a_dst, segmax, expw, denom);
    gat_edge_aggr<<<2048, T, 0, stream>>>(ei, expw, denom, hsrc, agg);
    gat_gemm<true><<<gemm_blocks, T, 0, stream>>>(agg, W_lin, bias, b_lin, out);
}